// QuantumOnly_21938692948015
// MI455X (gfx1250) — compile-verified
//
#include <hip/hip_runtime.h>
#include <cstdint>
#include <math.h>

// out[b] = fc_w * prod_{w=1..6} cos(x_q[b,w] + q_params[w]) + fc_b
// (closed form of the 7-qubit RY+CNOT-ring circuit's <Z_0>; wire 0 cancels,
//  x_c is unused by the reference)

#define NQ 7
#define TILE 256
#define TILE_BYTES (TILE * NQ * 4)   // 7168 bytes per block tile of x_q
#define CHUNKS (TILE_BYTES / 16)     // 448 x 16B async copies

__global__ __launch_bounds__(TILE) void quantum_expz_kernel(
    const float* __restrict__ x_q,
    const float* __restrict__ q_params,
    const float* __restrict__ fc_w,
    const float* __restrict__ fc_b,
    float* __restrict__ out,
    int B)
{
    __shared__ float tile[TILE * NQ];

    const int t  = threadIdx.x;
    const int b0 = blockIdx.x * TILE;
    const int b  = b0 + t;

    const bool full = (b0 + TILE) <= B;   // uniform per block

    if (full) {
        // Stage this block's 7168-byte x_q tile into LDS using gfx1250
        // async global->LDS copies (ASYNCcnt-tracked, no VGPR destinations).
        // LDS byte address = low 32 bits of the flat pointer to the LDS array.
        const uint32_t lds_base = (uint32_t)(uintptr_t)(&tile[0]);
        const uint32_t gbase    = (uint32_t)(b0 * (NQ * 4));  // byte offset, < 2^31

        {
            const uint32_t lo = lds_base + (uint32_t)t * 16u;
            const uint32_t go = gbase + (uint32_t)t * 16u;
            asm volatile("global_load_async_to_lds_b128 %0, %1, %2"
                         :
                         : "v"(lo), "v"(go), "s"(x_q)
                         : "memory");
        }
        if (t < (CHUNKS - TILE)) {        // remaining 192 chunks
            const uint32_t lo = lds_base + (uint32_t)(t + TILE) * 16u;
            const uint32_t go = gbase + (uint32_t)(t + TILE) * 16u;
            asm volatile("global_load_async_to_lds_b128 %0, %1, %2"
                         :
                         : "v"(lo), "v"(go), "s"(x_q)
                         : "memory");
        }
#if __has_builtin(__builtin_amdgcn_s_wait_asynccnt)
        __builtin_amdgcn_s_wait_asynccnt(0);
#else
        asm volatile("s_wait_asynccnt 0" ::: "memory");
#endif
    }
    __syncthreads();  // every wave waited on its own copies above, so after
                      // the barrier the whole tile is valid in LDS

    if (b < B) {
        // Uniform (scalar-promotable) parameter reads.
        const float w_scale = fc_w[0];
        const float w_bias  = fc_b[0];

        float prod = 1.0f;
        if (full) {
            const int base = t * NQ;
#pragma unroll
            for (int w = 1; w < NQ; ++w) {
                prod *= cosf(tile[base + w] + q_params[w]);
            }
        } else {
            const long long gb = (long long)b * NQ;
#pragma unroll
            for (int w = 1; w < NQ; ++w) {
                prod *= cosf(x_q[gb + w] + q_params[w]);
            }
        }
        out[b] = prod * w_scale + w_bias;
    }
}

extern "C" void kernel_launch(void* const* d_in, const int* in_sizes, int n_in,
                              void* d_out, int out_size, void* d_ws, size_t ws_size,
                              hipStream_t stream) {
    const float* x_q      = (const float*)d_in[0];
    // d_in[1] = x_c : unused by the reference computation
    const float* q_params = (const float*)d_in[2];
    const float* fc_w     = (const float*)d_in[3];
    const float* fc_b     = (const float*)d_in[4];
    float* out            = (float*)d_out;

    const int B = in_sizes[0] / NQ;          // 262144
    const int grid = (B + TILE - 1) / TILE;  // 1024 blocks of 256 threads (8 waves)

    quantum_expz_kernel<<<grid, TILE, 0, stream>>>(x_q, q_params, fc_w, fc_b, out, B);
}